// Huber_89816356094072
// MI455X (gfx1250) — compile-verified
//
#include <hip/hip_runtime.h>
#include <hip/hip_bf16.h>

#define C_HUB   1.345f
#define ALPHA_H 0.7102f
#define GAMMA_H 2.0f
#define MDIM    1024
#define NDIM    1024
#define RDIM    32

typedef __attribute__((ext_vector_type(2))) float v2f;
typedef __attribute__((ext_vector_type(8))) float v8f;

// ---------------------------------------------------------------------------
// Generic f32 WMMA GEMM:  C = (SUB ? Xsub - A@B : A@B)
// A is M x K (element A[m,k] = TA ? A[k*lda+m] : A[m*lda+k]), same for B.
// One wave (32 lanes) computes one 16x16 tile of C via V_WMMA_F32_16X16X4_F32.
// A frag (16x4): lanes 0-15 hold M=0..15 @ K=k0,k0+1 ; lanes 16-31 @ K=k0+2,k0+3
// B frag (4x16): lanes 0-15 hold N=0..15 @ K=k0,k0+1 ; lanes 16-31 @ K=k0+2,k0+3
// C frag: VGPR v -> (M = v + 8*(lane>>4), N = lane&15)
// Compile-time TA/TB/SUB -> straight-line K loop, 2 WMMAs per body.
// ---------------------------------------------------------------------------
template <bool TA, bool TB, bool SUB>
__global__ __launch_bounds__(32)
void wmma_gemm_f32(int Kdim,
                   const float* __restrict__ A, int lda,
                   const float* __restrict__ B, int ldb,
                   const float* __restrict__ Xsub,
                   float* __restrict__ Cm, int ldc) {
  const int lane = threadIdx.x & 31;
  const int m0 = blockIdx.y * 16;
  const int n0 = blockIdx.x * 16;
  const int half = lane >> 4;
  const int l15  = lane & 15;
  const int row  = m0 + l15;   // M index carried by this lane in A frag
  const int col  = n0 + l15;   // N index carried by this lane in B frag

  v8f acc = {};
#pragma unroll 2
  for (int k0 = 0; k0 < Kdim; k0 += 4) {
    const int ka = k0 + half * 2;
    v2f a, b;
    if (TA) { a.x = A[ka * lda + row];      a.y = A[(ka + 1) * lda + row]; }
    else    { a.x = A[row * lda + ka];      a.y = A[row * lda + ka + 1];   }
    if (TB) { b.x = B[col * ldb + ka];      b.y = B[col * ldb + ka + 1];   }
    else    { b.x = B[ka * ldb + col];      b.y = B[(ka + 1) * ldb + col]; }
    acc = __builtin_amdgcn_wmma_f32_16x16x4_f32(false, a, false, b,
                                                (short)0, acc, false, false);
  }
#pragma unroll
  for (int v = 0; v < 8; ++v) {
    const int m = m0 + v + half * 8;
    const int n = n0 + l15;
    const float val = acc[v];
    Cm[m * ldc + n] = SUB ? (Xsub[m * ldc + n] - val) : val;
  }
}

// ---------------------------------------------------------------------------
// 32x32 Gauss-Jordan inverse, one block of 256 threads.
// ---------------------------------------------------------------------------
__global__ void invert32_kernel(const float* __restrict__ G, float* __restrict__ Gi) {
  __shared__ float Mx[32][65];
  __shared__ float colk[32];
  const int t = threadIdx.x;
  for (int idx = t; idx < 32 * 64; idx += 256) {
    const int r = idx >> 6, c = idx & 63;
    Mx[r][c] = (c < 32) ? G[r * 32 + c] : ((c - 32) == r ? 1.0f : 0.0f);
  }
  __syncthreads();
  for (int k = 0; k < 32; ++k) {
    const float pinv = 1.0f / Mx[k][k];
    __syncthreads();
    if (t < 64) Mx[k][t] *= pinv;
    __syncthreads();
    if (t < 32) colk[t] = Mx[t][k];
    __syncthreads();
    for (int idx = t; idx < 32 * 64; idx += 256) {
      const int r = idx >> 6, c = idx & 63;
      if (r != k) Mx[r][c] -= colk[r] * Mx[k][c];
    }
    __syncthreads();
  }
  for (int idx = t; idx < 32 * 32; idx += 256) {
    const int r = idx >> 5, c = idx & 31;
    Gi[r * 32 + c] = Mx[r][32 + c];
  }
}

// ---------------------------------------------------------------------------
// Per-slice robust scale: scale[j] = 1.4815 * median(|r - median(r)|)
// slice j: elements Rm[j*sliceStride + k*elemStride], k = 0..1023.
// Block of 1024 threads; bitonic sort in LDS.
// ---------------------------------------------------------------------------
__device__ __forceinline__ void bitonic1024(float* buf) {
  const int i = threadIdx.x;
  for (int k = 2; k <= 1024; k <<= 1)
    for (int j = k >> 1; j > 0; j >>= 1) {
      __syncthreads();
      const int ixj = i ^ j;
      if (ixj > i) {
        const bool up = ((i & k) == 0);
        const float a = buf[i], b = buf[ixj];
        if ((a > b) == up) { buf[i] = b; buf[ixj] = a; }
      }
    }
  __syncthreads();
}

__global__ void median_scale_kernel(const float* __restrict__ Rm,
                                    float* __restrict__ scale,
                                    int elemStride, int sliceStride) {
  __shared__ float buf[1024];
  const int j = blockIdx.x, t = threadIdx.x;
  buf[t] = Rm[j * sliceStride + t * elemStride];
  bitonic1024(buf);
  const float med = 0.5f * (buf[511] + buf[512]);
  __syncthreads();
  buf[t] = fabsf(buf[t] - med);   // same multiset as |r - med|
  bitonic1024(buf);
  if (t == 0) scale[j] = 1.4815f * 0.5f * (buf[511] + buf[512]);
}

// ---------------------------------------------------------------------------
// scale[j] = sqrt(GAMMA*s^2/denom * sum_k 0.5*min((r/s)^2, C^2))
// ---------------------------------------------------------------------------
__global__ void chi_scale_kernel(const float* __restrict__ Rm,
                                 float* __restrict__ scale,
                                 int elemStride, int sliceStride, float denom) {
  __shared__ float red[256];
  const int j = blockIdx.x, t = threadIdx.x;
  const float s = scale[j];
  const float inv = 1.0f / s;
  const float c2 = C_HUB * C_HUB;
  float acc = 0.0f;
  for (int k = t; k < 1024; k += 256) {
    const float u = Rm[j * sliceStride + k * elemStride] * inv;
    acc += 0.5f * fminf(u * u, c2);
  }
  red[t] = acc; __syncthreads();
  for (int o = 128; o > 0; o >>= 1) { if (t < o) red[t] += red[t + o]; __syncthreads(); }
  if (t == 0) scale[j] = sqrtf(GAMMA_H * s * s / denom * red[0]);
}

// ---------------------------------------------------------------------------
// V-step beta update (one block per column j, 256 threads):
//   rp[w]   = clamp(R[w][j], -C*s, C*s)
//   S[h][d] = sum_w Up[h][w] * rp[w - d + 1]        (zero-padded)
//   V[h][j] += sum_{dh,dw} kv[j][dh][dw] * S[h+dh-1][dw]
// ---------------------------------------------------------------------------
__global__ void beta_update_V_kernel(const float* __restrict__ Rm,
                                     const float* __restrict__ scale,
                                     const float* __restrict__ Up,
                                     const float* __restrict__ ck,
                                     float* __restrict__ Vcur) {
  __shared__ float rp[1024];
  __shared__ float S[32][3];
  const int j = blockIdx.x, t = threadIdx.x;
  const float cs = C_HUB * scale[j];
  for (int k = t; k < 1024; k += 256) {
    const float r = Rm[k * NDIM + j];
    rp[k] = fminf(fmaxf(r, -cs), cs);
  }
  __syncthreads();
  const int wave = t >> 5, lane = t & 31;
  for (int h = wave; h < 32; h += 8) {
    float s0 = 0, s1 = 0, s2 = 0;
    const float* urow = Up + h * NDIM;
    for (int w = lane; w < 1024; w += 32) {
      const float uv = urow[w];
      s0 += uv * (w + 1 < 1024 ? rp[w + 1] : 0.0f);
      s1 += uv * rp[w];
      s2 += uv * (w > 0 ? rp[w - 1] : 0.0f);
    }
    for (int o = 16; o > 0; o >>= 1) {
      s0 += __shfl_down(s0, o, 32);
      s1 += __shfl_down(s1, o, 32);
      s2 += __shfl_down(s2, o, 32);
    }
    if (lane == 0) { S[h][0] = s0; S[h][1] = s1; S[h][2] = s2; }
  }
  __syncthreads();
  if (t < 32) {
    const float* kj = ck + j * 9;     // kv = conv_kernels[:N]
    float bu = 0.0f;
    for (int dh = 0; dh < 3; ++dh) {
      const int row = t + dh - 1;
      if (row < 0 || row >= 32) continue;
      for (int dw = 0; dw < 3; ++dw) bu += kj[dh * 3 + dw] * S[row][dw];
    }
    Vcur[t * NDIM + j] += bu;
  }
}

// ---------------------------------------------------------------------------
// U-step beta update (one block per row i, 256 threads):
//   rp[c]    = clamp(R[i][c], -C*s, C*s)
//   T[d][rr] = sum_c Vp[c][rr] * rp[c - d + 1]      (zero-padded)
//   U[i][r] += sum_{dh,dw} ku[i][dh][dw] * T[dh][r+dw-1]
// ---------------------------------------------------------------------------
__global__ void beta_update_U_kernel(const float* __restrict__ Rm,
                                     const float* __restrict__ scale,
                                     const float* __restrict__ Vp,
                                     const float* __restrict__ ck,
                                     float* __restrict__ Ucur) {
  __shared__ float rp[1024];
  __shared__ float T[3][32];
  const int i = blockIdx.x, t = threadIdx.x;
  const float cs = C_HUB * scale[i];
  for (int c = t; c < 1024; c += 256) {
    const float r = Rm[i * NDIM + c];
    rp[c] = fminf(fmaxf(r, -cs), cs);
  }
  __syncthreads();
  const int wave = t >> 5, lane = t & 31;
  for (int rr = wave; rr < 32; rr += 8) {
    float t0 = 0, t1 = 0, t2 = 0;
    for (int c = lane; c < 1024; c += 32) {
      const float v = Vp[c * RDIM + rr];
      t0 += v * (c + 1 < 1024 ? rp[c + 1] : 0.0f);
      t1 += v * rp[c];
      t2 += v * (c > 0 ? rp[c - 1] : 0.0f);
    }
    for (int o = 16; o > 0; o >>= 1) {
      t0 += __shfl_down(t0, o, 32);
      t1 += __shfl_down(t1, o, 32);
      t2 += __shfl_down(t2, o, 32);
    }
    if (lane == 0) { T[0][rr] = t0; T[1][rr] = t1; T[2][rr] = t2; }
  }
  __syncthreads();
  if (t < 32) {
    const float* ki = ck + (NDIM - 1 + i) * 9;   // ku = conv_kernels[N-1 : N-1+M]
    float bu = 0.0f;
    for (int dh = 0; dh < 3; ++dh)
      for (int dw = 0; dw < 3; ++dw) {
        const int rr = t + dw - 1;
        if (rr < 0 || rr >= 32) continue;
        bu += ki[dh * 3 + dw] * T[dh][rr];
      }
    Ucur[i * RDIM + t] += bu;
  }
}

// out (32 x 2048) = [U_new^T | V_new]
__global__ void write_out_kernel(const float* __restrict__ Ucur,
                                 const float* __restrict__ Vcur,
                                 float* __restrict__ out) {
  const int idx = blockIdx.x * 256 + threadIdx.x;
  if (idx >= 32 * 2048) return;
  const int r = idx / 2048, c = idx % 2048;
  out[idx] = (c < 1024) ? Ucur[c * RDIM + r] : Vcur[r * NDIM + (c - 1024)];
}

// ---------------------------------------------------------------------------
extern "C" void kernel_launch(void* const* d_in, const int* in_sizes, int n_in,
                              void* d_out, int out_size, void* d_ws, size_t ws_size,
                              hipStream_t stream) {
  const float* X  = (const float*)d_in[0];
  const float* U0 = (const float*)d_in[1];
  const float* V0 = (const float*)d_in[2];
  const float* ck = (const float*)d_in[3];
  float* out = (float*)d_out;

  float* ws    = (float*)d_ws;
  float* Rmat  = ws;                       // 1024*1024
  float* Ucur  = Rmat + MDIM * NDIM;       // 1024*32
  float* Vcur  = Ucur + MDIM * RDIM;       // 32*1024
  float* Up    = Vcur + RDIM * NDIM;       // 32*1024
  float* Vp    = Up   + RDIM * NDIM;       // 1024*32
  float* G     = Vp   + MDIM * RDIM;       // 32*32
  float* Gi    = G    + RDIM * RDIM;       // 32*32
  float* scale = Gi   + RDIM * RDIM;       // 1024

  hipMemcpyAsync(Ucur, U0, MDIM * RDIM * sizeof(float), hipMemcpyDeviceToDevice, stream);
  hipMemcpyAsync(Vcur, V0, RDIM * NDIM * sizeof(float), hipMemcpyDeviceToDevice, stream);

  const float denom = 2.0f * ALPHA_H * (1024.0f - 32.0f - 1.0f);
  const dim3 blk32(32);
  const dim3 gridRR(RDIM / 16, RDIM / 16);     // 32x32 outputs
  const dim3 gridRN(NDIM / 16, RDIM / 16);     // 32x1024
  const dim3 gridMR(RDIM / 16, MDIM / 16);     // 1024x32
  const dim3 gridMN(NDIM / 16, MDIM / 16);     // 1024x1024

  // ---------------- V step ----------------
  // G = U^T U ; Gi = G^-1 ; Up = Gi @ U^T
  wmma_gemm_f32<true, false, false><<<gridRR, blk32, 0, stream>>>(
      MDIM, Ucur, RDIM, Ucur, RDIM, nullptr, G, RDIM);
  invert32_kernel<<<1, 256, 0, stream>>>(G, Gi);
  wmma_gemm_f32<false, true, false><<<gridRN, blk32, 0, stream>>>(
      RDIM, Gi, RDIM, Ucur, RDIM, nullptr, Up, NDIM);

  // r0 and robust scale (per column)
  wmma_gemm_f32<false, false, true><<<gridMN, blk32, 0, stream>>>(
      RDIM, Ucur, RDIM, Vcur, NDIM, X, Rmat, NDIM);
  median_scale_kernel<<<NDIM, 1024, 0, stream>>>(Rmat, scale, /*elemStride=*/NDIM, /*sliceStride=*/1);

  for (int it = 0; it < 3; ++it) {
    wmma_gemm_f32<false, false, true><<<gridMN, blk32, 0, stream>>>(
        RDIM, Ucur, RDIM, Vcur, NDIM, X, Rmat, NDIM);
    chi_scale_kernel<<<NDIM, 256, 0, stream>>>(Rmat, scale, NDIM, 1, denom);
    beta_update_V_kernel<<<NDIM, 256, 0, stream>>>(Rmat, scale, Up, ck, Vcur);
  }

  // ---------------- U step ----------------
  // G = V V^T ; Gi = G^-1 ; Vp = V^T @ Gi
  wmma_gemm_f32<false, true, false><<<gridRR, blk32, 0, stream>>>(
      NDIM, Vcur, NDIM, Vcur, NDIM, nullptr, G, RDIM);
  invert32_kernel<<<1, 256, 0, stream>>>(G, Gi);
  wmma_gemm_f32<true, false, false><<<gridMR, blk32, 0, stream>>>(
      RDIM, Vcur, NDIM, Gi, RDIM, nullptr, Vp, RDIM);

  // r0 and robust scale (per row)
  wmma_gemm_f32<false, false, true><<<gridMN, blk32, 0, stream>>>(
      RDIM, Ucur, RDIM, Vcur, NDIM, X, Rmat, NDIM);
  median_scale_kernel<<<MDIM, 1024, 0, stream>>>(Rmat, scale, /*elemStride=*/1, /*sliceStride=*/NDIM);

  for (int it = 0; it < 3; ++it) {
    wmma_gemm_f32<false, false, true><<<gridMN, blk32, 0, stream>>>(
        RDIM, Ucur, RDIM, Vcur, NDIM, X, Rmat, NDIM);
    chi_scale_kernel<<<MDIM, 256, 0, stream>>>(Rmat, scale, 1, NDIM, denom);
    beta_update_U_kernel<<<MDIM, 256, 0, stream>>>(Rmat, scale, Vp, ck, Ucur);
  }

  write_out_kernel<<<(32 * 2048 + 255) / 256, 256, 0, stream>>>(Ucur, Vcur, out);
}